// MambaSequenceClassification_86234353369908
// MI455X (gfx1250) — compile-verified
//
#include <hip/hip_runtime.h>
#include <hip/hip_bf16.h>
#include <math.h>

// ---------------- model constants ----------------
#define BATCH   8
#define SEQ     2048
#define DM      256      // D_MODEL
#define DI      512      // D_INNER
#define NH      8        // NHEADS
#define HD      64       // HEADDIM
#define DS      16       // D_STATE
#define DCONV   4
#define CONVD   544      // D_INNER + 2*D_STATE
#define DIP     1064     // 2*DI + 2*DS + NH
#define NLAYERS 4
#define BL      (BATCH*SEQ)   // 16384 rows
#define EPSV    1e-5f

// chunked-scan geometry
#define GCHUNK  128
#define NCHUNK  (SEQ / GCHUNK)   // 16

typedef __attribute__((ext_vector_type(2))) float v2f;
typedef __attribute__((ext_vector_type(8))) float v8f;

__device__ __forceinline__ float siluf(float x) {
    return x / (1.0f + __expf(-x));
}
__device__ __forceinline__ float softplusf(float x) {
    return (x > 20.0f) ? x : log1pf(__expf(x));
}

// ---------------- 1) embedding + position + mask ----------------
__global__ __launch_bounds__(256)
void embed_kernel(const int* __restrict__ x, const float* __restrict__ mask,
                  const float* __restrict__ emb, const float* __restrict__ pos,
                  float* __restrict__ h) {
    int idx = blockIdx.x * blockDim.x + threadIdx.x;     // over BL*DM
    if (idx >= BL * DM) return;
    int d  = idx & (DM - 1);
    int bl = idx >> 8;           // idx / DM
    int l  = bl & (SEQ - 1);     // bl % SEQ
    int tok = x[bl];
    h[idx] = (emb[tok * DM + d] + pos[l * DM + d]) * mask[bl];
}

// ---------------- 2) WMMA GEMM: Y[M,N] = X[M,K] * W[N,K]^T ----------------
// One wave32 computes a 2x2 block of 16x16 tiles (32x32 of Y): per k-step it
// loads 2 A-fragments + 2 B-fragments and issues 4 V_WMMA_F32_16X16X4_F32
// (1.0 fragment-load per WMMA). M must be a multiple of 32, K of 4; the N
// tail runs a separate masked loop picked by a uniform scalar branch.
__global__ __launch_bounds__(32)
void gemm_xwT_wmma(const float* __restrict__ X, const float* __restrict__ W,
                   float* __restrict__ Y, int M, int N, int K) {
    const int gN   = blockIdx.x;          // 32-column group
    const int gM   = blockIdx.y;          // 32-row group
    const int lane = threadIdx.x;         // 0..31
    const int half = lane >> 4;           // 0: K+{0,1}, 1: K+{2,3}
    const int l16  = lane & 15;

    const int rowA0 = gM * 32 + l16;      // always valid (M % 32 == 0)
    const int rowA1 = rowA0 + 16;
    const float* __restrict__ X0 = X + (size_t)rowA0 * K;
    const float* __restrict__ X1 = X + (size_t)rowA1 * K;

    const int col0 = gN * 32 + l16;       // may be OOB in tail group
    const int col1 = col0 + 16;
    const int c0c  = (col0 < N) ? col0 : (N - 1);
    const int c1c  = (col1 < N) ? col1 : (N - 1);
    const float m0 = (col0 < N) ? 1.0f : 0.0f;
    const float m1 = (col1 < N) ? 1.0f : 0.0f;
    const float* __restrict__ W0 = W + (size_t)c0c * K;
    const float* __restrict__ W1 = W + (size_t)c1c * K;

    v8f acc00 = {}, acc01 = {}, acc10 = {}, acc11 = {};

    if (gN * 32 + 32 <= N) {              // uniform: full 32-column group
        for (int k = 0; k < K; k += 4) {
            const int kk = k + half * 2;
            v2f a0, a1, b0, b1;
            a0.x = X0[kk]; a0.y = X0[kk + 1];
            a1.x = X1[kk]; a1.y = X1[kk + 1];
            b0.x = W0[kk]; b0.y = W0[kk + 1];
            b1.x = W1[kk]; b1.y = W1[kk + 1];
            acc00 = __builtin_amdgcn_wmma_f32_16x16x4_f32(false, a0, false, b0, (short)0, acc00, false, false);
            acc01 = __builtin_amdgcn_wmma_f32_16x16x4_f32(false, a0, false, b1, (short)0, acc01, false, false);
            acc10 = __builtin_amdgcn_wmma_f32_16x16x4_f32(false, a1, false, b0, (short)0, acc10, false, false);
            acc11 = __builtin_amdgcn_wmma_f32_16x16x4_f32(false, a1, false, b1, (short)0, acc11, false, false);
        }
    } else {                              // tail group: mask OOB columns
        for (int k = 0; k < K; k += 4) {
            const int kk = k + half * 2;
            v2f a0, a1, b0, b1;
            a0.x = X0[kk];      a0.y = X0[kk + 1];
            a1.x = X1[kk];      a1.y = X1[kk + 1];
            b0.x = W0[kk] * m0; b0.y = W0[kk + 1] * m0;
            b1.x = W1[kk] * m1; b1.y = W1[kk + 1] * m1;
            acc00 = __builtin_amdgcn_wmma_f32_16x16x4_f32(false, a0, false, b0, (short)0, acc00, false, false);
            acc01 = __builtin_amdgcn_wmma_f32_16x16x4_f32(false, a0, false, b1, (short)0, acc01, false, false);
            acc10 = __builtin_amdgcn_wmma_f32_16x16x4_f32(false, a1, false, b0, (short)0, acc10, false, false);
            acc11 = __builtin_amdgcn_wmma_f32_16x16x4_f32(false, a1, false, b1, (short)0, acc11, false, false);
        }
    }

    // D layout: lane l16 carries column; VGPR r carries M = base + r + 8*half.
    #pragma unroll
    for (int r = 0; r < 8; ++r) {
        const int mr0 = gM * 32 + r + half * 8;   // tile-row 0
        const int mr1 = mr0 + 16;                 // tile-row 1
        if (col0 < N) {
            Y[(size_t)mr0 * N + col0] = acc00[r];
            Y[(size_t)mr1 * N + col0] = acc10[r];
        }
        if (col1 < N) {
            Y[(size_t)mr0 * N + col1] = acc01[r];
            Y[(size_t)mr1 * N + col1] = acc11[r];
        }
    }
}

// ---------------- 3a) causal conv1d(width 4) + SiLU ----------------
__global__ __launch_bounds__(256)
void conv_silu_kernel(const float* __restrict__ zxbcdt,
                      const float* __restrict__ cw, const float* __restrict__ cb,
                      float* __restrict__ xBC) {
    int idx = blockIdx.x * blockDim.x + threadIdx.x;    // over BL*CONVD
    if (idx >= BL * CONVD) return;
    int c  = idx % CONVD;
    int bl = idx / CONVD;
    int l  = bl & (SEQ - 1);
    int b  = bl >> 11;          // bl / SEQ
    float acc = cb[c];
    #pragma unroll
    for (int k = 0; k < DCONV; ++k) {
        int t = l + k - (DCONV - 1);
        if (t >= 0) {
            acc += zxbcdt[((size_t)(b * SEQ + t)) * DIP + DI + c] * cw[c * DCONV + k];
        }
    }
    xBC[idx] = siluf(acc);
}

// ---------------- 3b) dt = softplus(raw + bias) ----------------
__global__ __launch_bounds__(256)
void dt_kernel(const float* __restrict__ zxbcdt, const float* __restrict__ dtb,
               float* __restrict__ dt) {
    int idx = blockIdx.x * blockDim.x + threadIdx.x;    // over BL*NH
    if (idx >= BL * NH) return;
    int hh = idx & (NH - 1);
    int bl = idx >> 3;
    float v = zxbcdt[(size_t)bl * DIP + DI + CONVD + hh] + dtb[hh];
    dt[idx] = softplusf(v);
}

// ---------------- 4) chunked selective-state scan ----------------
// Recurrence h_t = dA_t*h_{t-1} + (dt_t x_t) B_t^T is linear in h, so:
//   pass1: per chunk, local end-state from h=0, plus decay product exp(A*sum dt)
//   pass2: 16-step scan over chunk states -> carry-in per chunk
//   pass3: replay each chunk from its carry-in, emitting y (+ D*x)
// State layout S/Hin: [(bh*NCHUNK + c)*HD + p]*DS + n.

// pass 1: grid = B*NH*NCHUNK blocks, 64 threads (p = headdim index)
__global__ __launch_bounds__(64)
void scan_pass1(const float* __restrict__ xBC, const float* __restrict__ dt,
                const float* __restrict__ Alog,
                float* __restrict__ S, float* __restrict__ P) {
    const int blk = blockIdx.x;
    const int c   = blk % NCHUNK;
    const int bh  = blk / NCHUNK;
    const int b   = bh / NH;
    const int hh  = bh % NH;
    const int p   = threadIdx.x;

    const float A = -__expf(Alog[hh]);

    float state[DS];
    #pragma unroll
    for (int n = 0; n < DS; ++n) state[n] = 0.0f;
    float sumdt = 0.0f;

    const int t0 = c * GCHUNK;
    for (int t = t0; t < t0 + GCHUNK; ++t) {
        const size_t bl = (size_t)b * SEQ + t;
        __builtin_prefetch(xBC + (bl + 1) * CONVD + DI, 0, 1);  // next B/C row
        const float dtv = dt[bl * NH + hh];
        const float dA  = __expf(dtv * A);
        const float xv  = xBC[bl * CONVD + hh * HD + p];
        const float dtx = dtv * xv;
        const float* __restrict__ Bp = xBC + bl * CONVD + DI;
        #pragma unroll
        for (int n = 0; n < DS; ++n) state[n] = dA * state[n] + dtx * Bp[n];
        sumdt += dtv;
    }

    const size_t base = ((size_t)blk * HD + p) * DS;
    #pragma unroll
    for (int n = 0; n < DS; ++n) S[base + n] = state[n];
    if (p == 0) P[blk] = __expf(A * sumdt);
}

// pass 2: grid = B*NH blocks, 64 threads; tiny 16-step scan over chunks
__global__ __launch_bounds__(64)
void scan_pass2(const float* __restrict__ S, const float* __restrict__ P,
                float* __restrict__ Hin) {
    const int bh = blockIdx.x;
    const int p  = threadIdx.x;
    float carry[DS];
    #pragma unroll
    for (int n = 0; n < DS; ++n) carry[n] = 0.0f;
    for (int c = 0; c < NCHUNK; ++c) {
        const int blk = bh * NCHUNK + c;
        const size_t base = ((size_t)blk * HD + p) * DS;
        const float pc = P[blk];
        #pragma unroll
        for (int n = 0; n < DS; ++n) {
            Hin[base + n] = carry[n];
            carry[n] = pc * carry[n] + S[base + n];
        }
    }
}

// pass 3: grid = B*NH*NCHUNK blocks, 64 threads; replay chunk from carry-in
__global__ __launch_bounds__(64)
void scan_pass3(const float* __restrict__ xBC, const float* __restrict__ dt,
                const float* __restrict__ Alog, const float* __restrict__ Dp,
                const float* __restrict__ Hin, float* __restrict__ y) {
    const int blk = blockIdx.x;
    const int c   = blk % NCHUNK;
    const int bh  = blk / NCHUNK;
    const int b   = bh / NH;
    const int hh  = bh % NH;
    const int p   = threadIdx.x;

    const float A     = -__expf(Alog[hh]);
    const float Dcoef = Dp[hh];

    float state[DS];
    const size_t base = ((size_t)blk * HD + p) * DS;
    #pragma unroll
    for (int n = 0; n < DS; ++n) state[n] = Hin[base + n];

    const int t0 = c * GCHUNK;
    for (int t = t0; t < t0 + GCHUNK; ++t) {
        const size_t bl = (size_t)b * SEQ + t;
        __builtin_prefetch(xBC + (bl + 1) * CONVD + DI, 0, 1);  // next B/C row
        const float dtv = dt[bl * NH + hh];
        const float dA  = __expf(dtv * A);
        const float xv  = xBC[bl * CONVD + hh * HD + p];
        const float dtx = dtv * xv;
        const float* __restrict__ Bp = xBC + bl * CONVD + DI;
        const float* __restrict__ Cp = Bp + DS;
        float yv = 0.0f;
        #pragma unroll
        for (int n = 0; n < DS; ++n) {
            state[n] = dA * state[n] + dtx * Bp[n];
            yv += state[n] * Cp[n];
        }
        y[bl * DI + hh * HD + p] = yv + Dcoef * xv;
    }
}

// ---------------- 5) gating (silu(z)) + RMSNorm over D_INNER ----------------
__global__ __launch_bounds__(512)
void gate_rms_kernel(const float* __restrict__ zxbcdt, const float* __restrict__ nw,
                     float* __restrict__ y) {
    const int bl = blockIdx.x;
    const int e  = threadIdx.x;            // 0..511
    const size_t yi = (size_t)bl * DI + e;
    const float z = zxbcdt[(size_t)bl * DIP + e];
    const float v = y[yi] * siluf(z);

    __shared__ float red[DI];
    red[e] = v * v;
    __syncthreads();
    for (int s = DI / 2; s > 0; s >>= 1) {
        if (e < s) red[e] += red[e + s];
        __syncthreads();
    }
    const float scale = rsqrtf(red[0] / (float)DI + EPSV);
    y[yi] = v * scale * nw[e];
}

// ---------------- 6) residual add + LayerNorm over D_MODEL ----------------
__global__ __launch_bounds__(256)
void resid_ln_kernel(const float* __restrict__ proj, const float* __restrict__ mask,
                     const float* __restrict__ lnw, const float* __restrict__ lnb,
                     float* __restrict__ h) {
    const int bl = blockIdx.x;
    const int d  = threadIdx.x;            // 0..255
    const size_t i = (size_t)bl * DM + d;
    const float v = h[i] + proj[i];

    __shared__ float red[DM];
    red[d] = v;
    __syncthreads();
    for (int s = DM / 2; s > 0; s >>= 1) {
        if (d < s) red[d] += red[d + s];
        __syncthreads();
    }
    const float mu = red[0] / (float)DM;
    __syncthreads();
    const float c = v - mu;
    red[d] = c * c;
    __syncthreads();
    for (int s = DM / 2; s > 0; s >>= 1) {
        if (d < s) red[d] += red[d + s];
        __syncthreads();
    }
    const float var = red[0] / (float)DM;
    h[i] = (c * rsqrtf(var + EPSV) * lnw[d] + lnb[d]) * mask[bl];
}

// ---------------- 7) mean-pool over L + classifier ----------------
__global__ __launch_bounds__(256)
void pool_cls_kernel(const float* __restrict__ h, const float* __restrict__ cw,
                     const float* __restrict__ cb, float* __restrict__ out) {
    const int b = blockIdx.x;
    const int d = threadIdx.x;
    float s = 0.0f;
    for (int l = 0; l < SEQ; ++l) s += h[((size_t)b * SEQ + l) * DM + d];
    __shared__ float pooled[DM];
    pooled[d] = s / (float)SEQ;
    __syncthreads();
    if (d < 2) {
        float acc = cb[d];
        for (int k = 0; k < DM; ++k) acc += pooled[k] * cw[d * DM + k];
        out[b * 2 + d] = acc;
    }
}

// ---------------- host orchestration ----------------
extern "C" void kernel_launch(void* const* d_in, const int* in_sizes, int n_in,
                              void* d_out, int out_size, void* d_ws, size_t ws_size,
                              hipStream_t stream) {
    const int*   x_tok  = (const int*)  d_in[0];
    const float* amask  = (const float*)d_in[1];
    const float* emb    = (const float*)d_in[2];
    const float* pos    = (const float*)d_in[3];
    const float* Wi_all = (const float*)d_in[4];   // (4, 1064, 256)
    const float* cw_all = (const float*)d_in[5];   // (4, 544, 4)
    const float* cb_all = (const float*)d_in[6];   // (4, 544)
    const float* dtb_all= (const float*)d_in[7];   // (4, 8)
    const float* Al_all = (const float*)d_in[8];   // (4, 8)
    const float* Dp_all = (const float*)d_in[9];   // (4, 8)
    const float* nw_all = (const float*)d_in[10];  // (4, 512)
    const float* Wo_all = (const float*)d_in[11];  // (4, 256, 512)
    const float* lnw    = (const float*)d_in[12];
    const float* lnb    = (const float*)d_in[13];
    const float* clsw   = (const float*)d_in[14];
    const float* clsb   = (const float*)d_in[15];
    float* out = (float*)d_out;

    // workspace carve-up (all regions fully written before read)
    char* ws = (char*)d_ws;
    float* buf_h   = (float*)ws;  ws += (size_t)BL * DM    * 4;                  // 16 MB
    float* buf_zx  = (float*)ws;  ws += (size_t)BL * DIP   * 4;                  // 69.7 MB
    float* buf_xbc = (float*)ws;  ws += (size_t)BL * CONVD * 4;                  // 35.6 MB
    float* buf_dt  = (float*)ws;  ws += (size_t)BL * NH    * 4;                  // 0.5 MB
    float* buf_y   = (float*)ws;  ws += (size_t)BL * DI    * 4;                  // 33.5 MB
    float* buf_pr  = (float*)ws;  ws += (size_t)BL * DM    * 4;                  // 16 MB
    float* buf_S   = (float*)ws;  ws += (size_t)BATCH*NH*NCHUNK * HD * DS * 4;   // 4 MB
    float* buf_Hin = (float*)ws;  ws += (size_t)BATCH*NH*NCHUNK * HD * DS * 4;   // 4 MB
    float* buf_P   = (float*)ws;  ws += (size_t)BATCH*NH*NCHUNK * 4;             // 4 KB

    // 1) embedding
    embed_kernel<<<(BL * DM + 255) / 256, 256, 0, stream>>>(x_tok, amask, emb, pos, buf_h);

    for (int i = 0; i < NLAYERS; ++i) {
        const float* Wi  = Wi_all  + (size_t)i * DIP * DM;
        const float* cw  = cw_all  + (size_t)i * CONVD * DCONV;
        const float* cb  = cb_all  + (size_t)i * CONVD;
        const float* dtb = dtb_all + (size_t)i * NH;
        const float* Al  = Al_all  + (size_t)i * NH;
        const float* Dpp = Dp_all  + (size_t)i * NH;
        const float* nw  = nw_all  + (size_t)i * DI;
        const float* Wo  = Wo_all  + (size_t)i * DM * DI;

        // 2) in_proj: (16384 x 256) x (1064 x 256)^T -> (16384 x 1064)
        {
            dim3 grid((DIP + 31) / 32, BL / 32);
            gemm_xwT_wmma<<<grid, 32, 0, stream>>>(buf_h, Wi, buf_zx, BL, DIP, DM);
        }
        // 3) conv + silu ; dt softplus
        conv_silu_kernel<<<(BL * CONVD + 255) / 256, 256, 0, stream>>>(buf_zx, cw, cb, buf_xbc);
        dt_kernel<<<(BL * NH + 255) / 256, 256, 0, stream>>>(buf_zx, dtb, buf_dt);
        // 4) chunked scan: 1024-way parallel, sequential depth ~2*128+16
        scan_pass1<<<BATCH * NH * NCHUNK, HD, 0, stream>>>(buf_xbc, buf_dt, Al, buf_S, buf_P);
        scan_pass2<<<BATCH * NH, HD, 0, stream>>>(buf_S, buf_P, buf_Hin);
        scan_pass3<<<BATCH * NH * NCHUNK, HD, 0, stream>>>(buf_xbc, buf_dt, Al, Dpp, buf_Hin, buf_y);
        // 5) gate + RMSNorm
        gate_rms_kernel<<<BL, DI, 0, stream>>>(buf_zx, nw, buf_y);
        // 6) out_proj: (16384 x 512) x (256 x 512)^T -> (16384 x 256)
        {
            dim3 grid((DM + 31) / 32, BL / 32);
            gemm_xwT_wmma<<<grid, 32, 0, stream>>>(buf_y, Wo, buf_pr, BL, DM, DI);
        }
        // residual + LayerNorm + mask (writes buf_h in place)
        resid_ln_kernel<<<BL, DM, 0, stream>>>(buf_pr, amask, lnw, lnb, buf_h);
    }

    // 7) pool + classifier
    pool_cls_kernel<<<BATCH, DM, 0, stream>>>(buf_h, clsw, clsb, out);
}